// GCN_Low_32873679684168
// MI455X (gfx1250) — compile-verified
//
#include <hip/hip_runtime.h>

#define D 128          // feature/embedding width (reference constant)
#define LDS_STRIDE 132 // 128 + 4 pad: 16B-aligned rows, bank-conflict-free column reads

typedef __attribute__((ext_vector_type(2))) float v2f;
typedef __attribute__((ext_vector_type(8))) float v8f;

// ---------------------------------------------------------------------------
// Kernel 1: out = 0.5 * feature   (vectorized float4, fully coalesced)
// ---------------------------------------------------------------------------
__global__ void gcn_init(const float* __restrict__ feat,
                         float* __restrict__ out, int total4) {
    int i = blockIdx.x * blockDim.x + threadIdx.x;
    if (i < total4) {
        const float4* f4 = (const float4*)feat;
        float4* o4 = (float4*)out;
        float4 v = f4[i];
        v.x *= 0.5f; v.y *= 0.5f; v.z *= 0.5f; v.w *= 0.5f;
        o4[i] = v;
    }
}

// ---------------------------------------------------------------------------
// Kernel 2: out[dst] += 0.5 * w * feature[src]
// One wave32 per edge: each lane moves 4 contiguous floats (512B/edge,
// coalesced) and issues 4 native f32 atomics (no-return -> STOREcnt path).
// ---------------------------------------------------------------------------
__global__ void gcn_scatter(const float* __restrict__ feat,
                            const int* __restrict__ esrc,
                            const int* __restrict__ edst,
                            const float* __restrict__ ew,
                            float* __restrict__ out, int n_edges) {
    int e    = (blockIdx.x * blockDim.x + threadIdx.x) >> 5;
    int lane = threadIdx.x & 31;
    if (e >= n_edges) return;
    int   src = esrc[e];
    int   dst = edst[e];
    float w   = 0.5f * ew[e];
    const float4* f4 = (const float4*)(feat + (size_t)src * D);
    float4 v = f4[lane];
    float* o = out + (size_t)dst * D + lane * 4;
    unsafeAtomicAdd(o + 0, w * v.x);
    unsafeAtomicAdd(o + 1, w * v.y);
    unsafeAtomicAdd(o + 2, w * v.z);
    unsafeAtomicAdd(o + 3, w * v.w);
}

// ---------------------------------------------------------------------------
// Kernel 3: out = blended @ W, in place. Block = 256 threads = 8 waves.
// Block b owns rows [16b, 16b+16): stage 16x128 A tile into LDS, then each
// wave computes one 16x16 output tile with V_WMMA_F32_16X16X4_F32 over K=128.
// N_NODES = 50000 is divisible by 16, so EXEC stays all-ones (WMMA requires).
// ---------------------------------------------------------------------------
__global__ void __launch_bounds__(256)
gcn_gemm(const float* __restrict__ weight, float* __restrict__ out) {
    __shared__ float As[16 * LDS_STRIDE];
    const int m0  = blockIdx.x * 16;
    const int tid = threadIdx.x;

    // Stage blended rows (currently living in `out`) into LDS.
    // 16*128 = 2048 floats = 512 float4; 256 threads x 2 float4 each.
    {
        const float4* src4 = (const float4*)(out + (size_t)m0 * D);
        #pragma unroll
        for (int j = 0; j < 2; ++j) {
            int idx = tid + j * 256;   // float4 index in [0,512)
            int row = idx >> 5;        // 32 float4 per row
            int c4  = idx & 31;
            float4 v = src4[idx];
            *(float4*)&As[row * LDS_STRIDE + c4 * 4] = v;  // 16B aligned
        }
    }
    __syncthreads();

    const int wave = tid >> 5;      // 0..7 -> n-tile
    const int lane = tid & 31;
    const int n0   = wave * 16;
    const int half = lane >> 4;     // K sub-pair select (ISA A/B layout)
    const int l16  = lane & 15;     // M (for A) / N (for B)

    v8f c = {0.f, 0.f, 0.f, 0.f, 0.f, 0.f, 0.f, 0.f};

    #pragma unroll 8
    for (int ki = 0; ki < 32; ++ki) {
        const int kbase = ki * 4 + half * 2;
        // A 16x4 f32: lane L holds A[L%16][kbase], A[L%16][kbase+1]
        const float* ap = &As[l16 * LDS_STRIDE + kbase];   // 8B-aligned -> ds_load_b64
        v2f a; a.x = ap[0]; a.y = ap[1];
        // B 4x16 f32: lane L holds B[kbase][n0+L%16], B[kbase+1][n0+L%16]
        const float* bp = weight + (size_t)kbase * D + (n0 + l16);
        v2f b; b.x = bp[0]; b.y = bp[D];
        c = __builtin_amdgcn_wmma_f32_16x16x4_f32(
                /*neg_a=*/false, a, /*neg_b=*/false, b,
                /*c_mod=*/(short)0, c, /*reuse_a=*/false, /*reuse_b=*/false);
    }

    // D 16x16 f32: VGPR r, lane L -> out[m0 + r + 8*(L/16)][n0 + L%16]
    #pragma unroll
    for (int r = 0; r < 8; ++r) {
        int row = m0 + r + half * 8;
        out[(size_t)row * D + n0 + l16] = c[r];
    }
}

// ---------------------------------------------------------------------------
extern "C" void kernel_launch(void* const* d_in, const int* in_sizes, int n_in,
                              void* d_out, int out_size, void* d_ws, size_t ws_size,
                              hipStream_t stream) {
    const float* feature = (const float*)d_in[0];
    const int*   esrc    = (const int*)  d_in[1];
    const int*   edst    = (const int*)  d_in[2];
    const float* ew      = (const float*)d_in[3];
    const float* weight  = (const float*)d_in[4];
    float*       out     = (float*)d_out;

    const int n_nodes = in_sizes[0] / D;   // 50000
    const int n_edges = in_sizes[1];       // 800000

    // 1) out = 0.5*feature
    const int total4 = n_nodes * (D / 4);
    gcn_init<<<(total4 + 255) / 256, 256, 0, stream>>>(feature, out, total4);

    // 2) out[dst] += 0.5*w*feature[src]   (8 edges per 256-thread block)
    const int epb = 256 / 32;
    gcn_scatter<<<(n_edges + epb - 1) / epb, 256, 0, stream>>>(
        feature, esrc, edst, ew, out, n_edges);

    // 3) out = out @ W  (in place; rows owned exclusively per block)
    gcn_gemm<<<n_nodes / 16, 256, 0, stream>>>(weight, out);
}